// CategoryDense_15822659519008
// MI455X (gfx1250) — compile-verified
//
#include <hip/hip_runtime.h>
#include <stdint.h>

typedef __attribute__((ext_vector_type(2))) float v2f;
typedef __attribute__((ext_vector_type(4))) float v4f;
typedef __attribute__((ext_vector_type(8))) float v8f;
typedef __attribute__((ext_vector_type(4))) unsigned int v4u;
typedef __attribute__((ext_vector_type(4))) int v4i;
typedef __attribute__((ext_vector_type(8))) int v8i;

#define B_SZ   8192
#define C_SZ   64
#define IN_SZ  64
#define OUT_SZ 64

#define THREADS        128
#define ROWS_PER_ITER  64                        // 4 waves * 16 rows
#define ITERS          4
#define ROWS_PER_BLOCK (ROWS_PER_ITER * ITERS)   // 256
#define MSLABS         (B_SZ / ROWS_PER_BLOCK)   // 32
#define XP             68   // padded LDS row stride for x tile (floats)
#define KP             68   // padded LDS row stride for transposed kernel (floats)

// Issue one TDM descriptor: load a 64-row x 64-col f32 tile of x (row stride
// C*IN floats in memory) into LDS with 4-DWORD padding after every 64 DWORDs,
// i.e. LDS row stride = 68 floats = XP.  Descriptor layout per CDNA5 ISA §8.3/8.4.
__device__ __forceinline__ void tdm_load_x_tile(const float* x, int c, int m_base,
                                                uint32_t lds_byte_addr) {
    const uint64_t ga = (uint64_t)(uintptr_t)(x + ((size_t)m_base * C_SZ + c) * IN_SZ);
    v4u g0;
    g0.x = 1u;                                        // count=1, user descriptor
    g0.y = lds_byte_addr;                             // LDS dest (bytes)
    g0.z = (uint32_t)(ga & 0xFFFFFFFFu);              // global_addr[31:0]
    g0.w = (uint32_t)((ga >> 32) & 0x01FFFFFFu)       // global_addr[56:32]
         | (2u << 30);                                // type = 2 ("image")
    v8i g1;
    g1[0] = (2 << 16)                                 // data_size = 4 bytes
          | (1 << 20)                                 // pad_enable
          | (5 << 22)                                 // pad_interval: 64 DWORDs
          | (3 << 25);                                // pad_amount:   4 DWORDs
    g1[1] = (int)((uint32_t)IN_SZ << 16);             // tensor_dim0 = 64   (bits 79:48)
    g1[2] = (int)((uint32_t)B_SZ  << 16);             // tensor_dim1 = 8192 (bits 111:80)
    g1[3] = (int)((uint32_t)IN_SZ << 16);             // tile_dim0  = 64    (bits 127:112)
    g1[4] = ROWS_PER_ITER;                            // tile_dim1  = 64    (bits 143:128)
    g1[5] = (int)(C_SZ * IN_SZ);                      // tensor_dim0_stride = 4096 (bits 207:160)
    g1[6] = 0;
    g1[7] = 0;
    const v4i gz4 = {0, 0, 0, 0};                     // groups 2/3 unused (2D tensor)
    const v8i gz8 = {0, 0, 0, 0, 0, 0, 0, 0};         // extra group (6-arg toolchain form)
    __builtin_amdgcn_tensor_load_to_lds(g0, g1, gz4, gz4, gz8, 0);
}

// One block: one category c, a 256-row slab of the batch, processed as four
// 64-row tiles.  x tiles are DMA'd into double-buffered LDS by the Tensor Data
// Mover (overlapped with compute); each wave computes a 16x64 output tile with
// v_wmma_f32_16x16x4_f32, accumulating over K=64 in steps of 4.
__global__ __launch_bounds__(THREADS)
void category_dense_wmma(const float* __restrict__ x,
                         const float* __restrict__ kern,
                         const float* __restrict__ bias,
                         float* __restrict__ out) {
    __shared__ float lds_kT[OUT_SZ * KP];            // kernel[c] transposed: [n][k]
    __shared__ float lds_b[OUT_SZ];                  // bias[c]
    __shared__ float lds_x[2][ROWS_PER_ITER * XP];   // double-buffered x tiles

    const int t     = threadIdx.x;
    const int c     = blockIdx.y;
    const int mslab = blockIdx.x;

    // ---- stage kernel[c] transposed: lds_kT[n*KP + k] = kern[c][k][n] ----
    {
        const v4f* gk = reinterpret_cast<const v4f*>(kern + (size_t)c * (IN_SZ * OUT_SZ));
#pragma unroll
        for (int j = 0; j < 8; ++j) {        // 1024 v4f / 128 threads
            const int f4 = j * THREADS + t;  // 0..1023
            const int k  = f4 >> 4;          // K row 0..63
            const int n0 = (f4 & 15) * 4;    // first of 4 N columns
            const v4f r  = gk[f4];
            lds_kT[(n0 + 0) * KP + k] = r.x;
            lds_kT[(n0 + 1) * KP + k] = r.y;
            lds_kT[(n0 + 2) * KP + k] = r.z;
            lds_kT[(n0 + 3) * KP + k] = r.w;
        }
    }
    if (t < OUT_SZ) lds_b[t] = bias[c * OUT_SZ + t];

    const int w    = t >> 5;   // wave id 0..3
    const int l    = t & 31;
    const int half = l >> 4;   // 0: lanes 0-15, 1: lanes 16-31
    const int l16  = l & 15;

    // prologue: kick off DMA of tile 0 (wave 0 issues; EXEC-independent op)
    if (t < 32) {
        tdm_load_x_tile(x, c, mslab * ROWS_PER_BLOCK,
                        (uint32_t)(uintptr_t)&lds_x[0][0]);
    }

    for (int it = 0; it < ITERS; ++it) {
        const int m_base = mslab * ROWS_PER_BLOCK + it * ROWS_PER_ITER;

        // tile `it` DMA complete (TDM ops complete in order; only one in flight)
        if (t < 32) __builtin_amdgcn_s_wait_tensorcnt(0);
        // makes tile `it` visible to all waves AND guarantees everyone finished
        // reading buffer (it+1)&1 (used by tile it-1) before we overwrite it
        __syncthreads();
        if (t < 32 && (it + 1) < ITERS) {
            tdm_load_x_tile(x, c, m_base + ROWS_PER_ITER,
                            (uint32_t)(uintptr_t)&lds_x[(it + 1) & 1][0]);
        }

        const float* xb  = lds_x[it & 1];
        const int mrow   = w * 16 + l16;   // A-fragment row (M = lane%16)

        v8f acc0, acc1, acc2, acc3;
        {
            // C/D element (vgpr v, lane l) = (M = v + 8*half, N = l16 + 16*nt):
            // bias depends only on N -> broadcast per accumulator.
            const float b0 = lds_b[ 0 + l16];
            const float b1 = lds_b[16 + l16];
            const float b2 = lds_b[32 + l16];
            const float b3 = lds_b[48 + l16];
#pragma unroll
            for (int v = 0; v < 8; ++v) {
                acc0[v] = b0; acc1[v] = b1; acc2[v] = b2; acc3[v] = b3;
            }
        }

#pragma unroll
        for (int k = 0; k < IN_SZ; k += 4) {
            const int kk = k + half * 2;   // A/B vgpr0 holds K = 2*half, vgpr1 K = 2*half+1

            // A fragment 16x4 f32: one aligned b64 LDS load
            const v2f a = *reinterpret_cast<const v2f*>(&xb[mrow * XP + kk]);
            // B fragments 4x16 f32 from transposed kernel: aligned b64 loads
            const v2f bf0 = *reinterpret_cast<const v2f*>(&lds_kT[( 0 + l16) * KP + kk]);
            const v2f bf1 = *reinterpret_cast<const v2f*>(&lds_kT[(16 + l16) * KP + kk]);
            const v2f bf2 = *reinterpret_cast<const v2f*>(&lds_kT[(32 + l16) * KP + kk]);
            const v2f bf3 = *reinterpret_cast<const v2f*>(&lds_kT[(48 + l16) * KP + kk]);

            acc0 = __builtin_amdgcn_wmma_f32_16x16x4_f32(false, a, false, bf0,
                                                         (short)0, acc0, false, false);
            acc1 = __builtin_amdgcn_wmma_f32_16x16x4_f32(false, a, false, bf1,
                                                         (short)0, acc1, false, false);
            acc2 = __builtin_amdgcn_wmma_f32_16x16x4_f32(false, a, false, bf2,
                                                         (short)0, acc2, false, false);
            acc3 = __builtin_amdgcn_wmma_f32_16x16x4_f32(false, a, false, bf3,
                                                         (short)0, acc3, false, false);
        }

        // ---- store 16x64 tile: row M = v + 8*half, col N = nt*16 + l16 ----
#pragma unroll
        for (int v = 0; v < 8; ++v) {
            const int grow = m_base + w * 16 + half * 8 + v;
            float* o = out + ((size_t)grow * C_SZ + c) * OUT_SZ;
            o[ 0 + l16] = acc0[v];
            o[16 + l16] = acc1[v];
            o[32 + l16] = acc2[v];
            o[48 + l16] = acc3[v];
        }
    }
}

extern "C" void kernel_launch(void* const* d_in, const int* in_sizes, int n_in,
                              void* d_out, int out_size, void* d_ws, size_t ws_size,
                              hipStream_t stream) {
    (void)in_sizes; (void)n_in; (void)out_size; (void)d_ws; (void)ws_size;
    const float* x    = (const float*)d_in[0];  // (8192, 64, 64)
    const float* kern = (const float*)d_in[1];  // (1, 64, 64, 64)
    const float* bias = (const float*)d_in[2];  // (1, 64, 64)
    float* out = (float*)d_out;                 // (8192, 64, 64)

    dim3 grid(MSLABS, C_SZ);  // 32 x 64 = 2048 blocks
    category_dense_wmma<<<grid, THREADS, 0, stream>>>(x, kern, bias, out);
}